// scattering_2d_2layers_21174188769982
// MI455X (gfx1250) — compile-verified
//
#include <hip/hip_runtime.h>
#include <hip/hip_bf16.h>
#include <math.h>

// ---------------------------------------------------------------------------
// 2D scattering transform, 2 layers.  N=512, K=8, J=6 (48 filters).
// Pipeline per filter f:
//   R1: row-wise  ifft( x_hat * psi_hat[f] )          (scale 1/512)
//   C : col-wise  ifft  -> modulus(+EPS) -> col fft   (scale 1/512 on ifft)
//   R2: row-wise  fft   -> power = |.|^2
// Reductions:
//   s0  = INV * mean(|x|^2 * |phi|^2)
//   a_f = INV * mean(power_f * |phi|^2)
//   B   = INV/NN * power[48xNN] @ psi2[NNx48]   <-- V_WMMA_F32_16X16X4_F32
// CDNA5 paths used: v_wmma_f32_16x16x4_f32, global_load_async_to_lds_b64 /
// global_store_async_from_lds_b64 (+ s_wait_asynccnt), global_prefetch,
// f32 global atomics.
// ---------------------------------------------------------------------------

#define NSZ   512
#define NN    (512 * 512)
#define NFILT 48
#define INV_NORM 0.025330295910584444f   /* 1/(2*pi)^2 */
#define EPSV 1e-8f

typedef __attribute__((ext_vector_type(2))) float v2f;
typedef __attribute__((ext_vector_type(8))) float v8f;

__device__ __forceinline__ int brev9(int e) {
    return (int)(__brev((unsigned)e) >> 23);   // 9-bit bit reversal
}

// Low 32 bits of a flat pointer into __shared__ are the LDS byte offset
// (see ISA 10.2).
__device__ __forceinline__ unsigned lds_off(const void* p) {
    return (unsigned)(size_t)p;
}

// Per-lane async DMA: copies 8 bytes from global address g to LDS offset loff.  Tracked by ASYNCcnt.
__device__ __forceinline__ void async_ld_b64(unsigned loff, const void* g) {
    asm volatile("global_load_async_to_lds_b64 %0, %1, off"
                 :: "v"(loff), "v"((unsigned long long)g) : "memory");
}
// Per-lane async DMA: copies 8 bytes from LDS offset loff to global address g.
__device__ __forceinline__ void async_st_b64(void* g, unsigned loff) {
    asm volatile("global_store_async_from_lds_b64 %0, %1, off"
                 :: "v"((unsigned long long)g), "v"(loff) : "memory");
}
__device__ __forceinline__ void wait_async0() {
    asm volatile("s_wait_asynccnt 0" ::: "memory");
}

// 512-point in-place radix-2 DIT FFT in LDS.  256 threads, input must already
// be in bit-reversed order.  sign=+1 -> inverse kernel (no 1/N scale applied
// here), sign=-1 -> forward.
__device__ __forceinline__ void fft512_lds(float2* s, int tid, float sign) {
    #pragma unroll
    for (int st = 0; st < 9; ++st) {
        int half = 1 << st;
        int grp  = tid >> st;
        int pos  = tid & (half - 1);
        int i0   = (grp << (st + 1)) + pos;
        int i1   = i0 + half;
        float ang = sign * 6.2831853071795864f * (float)pos / (float)(2 << st);
        float sn, cs;
        __sincosf(ang, &sn, &cs);
        __syncthreads();
        float2 u = s[i0];
        float2 v = s[i1];
        float tr = v.x * cs - v.y * sn;
        float ti = v.x * sn + v.y * cs;
        s[i0] = make_float2(u.x + tr, u.y + ti);
        s[i1] = make_float2(u.x - tr, u.y - ti);
    }
    __syncthreads();
}

// ---- pass 1: per (filter,row): complex-mul with x_hat, row ifft -------------
// (multiply must pass through VALU, so regular loads here)
__global__ void k_row_ifft_mul(const float* __restrict__ x_hat,
                               const float* __restrict__ psi_r,
                               const float* __restrict__ psi_i,
                               float2* __restrict__ buf) {
    __shared__ float2 s[NSZ];
    int tid = threadIdx.x;
    int row = blockIdx.x;
    int f   = blockIdx.y;
    const size_t base = (size_t)f * NN + (size_t)row * NSZ;
    const size_t xrow = (size_t)row * NSZ;
    for (int e = tid; e < NSZ; e += 256) {
        int src = brev9(e);
        float xr = x_hat[(xrow + src) * 2 + 0];
        float xi = x_hat[(xrow + src) * 2 + 1];
        float pr = psi_r[base + src];
        float pi = psi_i[base + src];
        s[e] = make_float2(xr * pr - xi * pi, xr * pi + xi * pr);
    }
    fft512_lds(s, tid, +1.0f);
    const float sc = 1.0f / 512.0f;
    for (int e = tid; e < NSZ; e += 256) {
        float2 v = s[e];
        buf[base + e] = make_float2(v.x * sc, v.y * sc);
    }
}

// ---- pass 2: per (filter,col): col ifft -> modulus -> col fft ---------------
// Column gather via async global->LDS DMA with the DIT bit-reversal folded
// into the per-lane LDS destination; write-back via async LDS->global DMA.
__global__ void k_col_ifft_mod_fft(float2* __restrict__ buf) {
    __shared__ float2 s[NSZ];
    __shared__ float  modv[NSZ];
    int tid = threadIdx.x;
    int col = blockIdx.x;
    int f   = blockIdx.y;
    const size_t base = (size_t)f * NN;
    for (int e = tid; e < NSZ; e += 256) {
        // s[brev(e)] = buf[f, e, col]   (brev involution == s[e]=buf[brev(e)])
        async_ld_b64(lds_off(&s[brev9(e)]), &buf[base + (size_t)e * NSZ + col]);
    }
    wait_async0();
    __syncthreads();
    fft512_lds(s, tid, +1.0f);
    const float sc = 1.0f / 512.0f;
    for (int e = tid; e < NSZ; e += 256) {
        float2 v = s[e];
        float re = v.x * sc, im = v.y * sc;
        modv[e] = sqrtf(re * re + im * im + EPSV);
    }
    __syncthreads();
    for (int e = tid; e < NSZ; e += 256) {
        s[e] = make_float2(modv[brev9(e)], 0.0f);
    }
    fft512_lds(s, tid, -1.0f);
    for (int e = tid; e < NSZ; e += 256) {
        async_st_b64(&buf[base + (size_t)e * NSZ + col], lds_off(&s[e]));
    }
    // S_ENDPGM implicitly waits for ASYNCcnt to drain.
}

// ---- pass 3: per (filter,row): row fft -> power -----------------------------
// Coalesced row read via async global->LDS DMA, bit-reversal on the LDS side.
__global__ void k_row_fft_power(const float2* __restrict__ buf,
                                float* __restrict__ power) {
    __shared__ float2 s[NSZ];
    int tid = threadIdx.x;
    int row = blockIdx.x;
    int f   = blockIdx.y;
    const size_t base = (size_t)f * NN + (size_t)row * NSZ;
    for (int e = tid; e < NSZ; e += 256) {
        async_ld_b64(lds_off(&s[brev9(e)]), &buf[base + e]);
    }
    wait_async0();
    __syncthreads();
    fft512_lds(s, tid, -1.0f);
    for (int e = tid; e < NSZ; e += 256) {
        float2 v = s[e];
        power[base + e] = v.x * v.x + v.y * v.y;
    }
}

// ---- s0 reduction -----------------------------------------------------------
__global__ void k_s0(const float* __restrict__ x_hat,
                     const float* __restrict__ phi_r,
                     const float* __restrict__ phi_i,
                     float* __restrict__ s0_ws) {
    __shared__ float red[256];
    int tid = threadIdx.x;
    float acc = 0.0f;
    for (size_t n = tid; n < (size_t)NN; n += 256) {
        float xr = x_hat[n * 2 + 0], xi = x_hat[n * 2 + 1];
        float pr = phi_r[n], pi = phi_i[n];
        acc += (xr * xr + xi * xi) * (pr * pr + pi * pi);
    }
    red[tid] = acc;
    __syncthreads();
    for (int st = 128; st > 0; st >>= 1) {
        if (tid < st) red[tid] += red[tid + st];
        __syncthreads();
    }
    if (tid == 0) s0_ws[0] = INV_NORM * red[0] / (float)NN;
}

// ---- a[f] = INV/NN * <power_f, |phi|^2> ------------------------------------
__global__ void k_a(const float* __restrict__ power,
                    const float* __restrict__ phi_r,
                    const float* __restrict__ phi_i,
                    float* __restrict__ a_ws) {
    __shared__ float red[256];
    int tid = threadIdx.x;
    int f   = blockIdx.x;
    const float* p = power + (size_t)f * NN;
    float acc = 0.0f;
    for (size_t n = tid; n < (size_t)NN; n += 256) {
        float pr = phi_r[n], pi = phi_i[n];
        acc += p[n] * (pr * pr + pi * pi);
    }
    red[tid] = acc;
    __syncthreads();
    for (int st = 128; st > 0; st >>= 1) {
        if (tid < st) red[tid] += red[tid + st];
        __syncthreads();
    }
    if (tid == 0) a_ws[f] = INV_NORM * red[0] / (float)NN;
}

// ---- B = INV/NN * power @ psi2^T via V_WMMA_F32_16X16X4_F32 -----------------
// One wave per (16x16 output tile, K-segment).  48x48 output = 3x3 tiles.
// NN split into 256 segments of 1024 -> 256 wmma ops per wave, f32 atomic
// accumulation of partial tiles.
__global__ void __launch_bounds__(32)
k_wmma_B(const float* __restrict__ power,
         const float* __restrict__ psi_r,
         const float* __restrict__ psi_i,
         float* __restrict__ Bacc) {
    const int lane = threadIdx.x;       // 0..31, wave32, EXEC all ones
    const int half = lane >> 4;
    const int l15  = lane & 15;
    const int tm   = blockIdx.x % 3;
    const int tn   = blockIdx.x / 3;
    const size_t seg   = blockIdx.y;
    const size_t chunk = NN / 256;      // 1024
    const size_t n0    = seg * chunk;
    const size_t rowA  = (size_t)(tm * 16 + l15) * NN;   // power row for A
    const size_t rowB  = (size_t)(tn * 16 + l15) * NN;   // psi2 row  for B
    const int kb = half * 2;            // lanes 0-15 hold K=0,1; 16-31 hold K=2,3

    v8f c = {};
    for (size_t n = n0; n < n0 + chunk; n += 4) {
        // pull next cachelines toward the WGP (global_prefetch)
        __builtin_prefetch(&power[rowA + n + 128], 0, 1);
        __builtin_prefetch(&psi_r[rowB + n + 128], 0, 1);
        v2f a, b;
        // A (16x4): a.x -> K=kb, a.y -> K=kb+1
        a.x = power[rowA + n + kb];
        a.y = power[rowA + n + kb + 1];
        // B (4x16): element [K][col] = psi2[tileN*16+col][n+K], on the fly
        float r0 = psi_r[rowB + n + kb],     i0 = psi_i[rowB + n + kb];
        float r1 = psi_r[rowB + n + kb + 1], i1 = psi_i[rowB + n + kb + 1];
        b.x = r0 * r0 + i0 * i0;
        b.y = r1 * r1 + i1 * i1;
        c = __builtin_amdgcn_wmma_f32_16x16x4_f32(
                /*neg_a=*/false, a, /*neg_b=*/false, b,
                /*c_mod=*/(short)0, c, /*reuse_a=*/false, /*reuse_b=*/false);
    }

    const float scale = INV_NORM / (float)NN;
    #pragma unroll
    for (int v = 0; v < 8; ++v) {
        int row = tm * 16 + v + half * 8;   // C/D layout: VGPR v -> M = v + 8*half
        int col = tn * 16 + l15;
        atomicAdd(&Bacc[row * NFILT + col], c[v] * scale);
    }
}

// ---- zero-init --------------------------------------------------------------
__global__ void k_zero(float* __restrict__ p, int n) {
    int i = blockIdx.x * blockDim.x + threadIdx.x;
    if (i < n) p[i] = 0.0f;
}

// ---- final output assembly (torch concatenation order) ----------------------
__global__ void k_assemble(const float* __restrict__ s0_ws,
                           const float* __restrict__ a_ws,
                           const float* __restrict__ Bacc,
                           float* __restrict__ out) {
    if (threadIdx.x != 0 || blockIdx.x != 0) return;
    int idx = 0;
    out[idx++] = s0_ws[0];
    for (int i = 0; i < 6; ++i) {
        for (int k = 0; k < 8; ++k) out[idx++] = a_ws[k * 6 + i];
        if (i < 5) {
            for (int k = 0; k < 8; ++k)
                for (int m = 0; m < 8; ++m)
                    for (int j = i + 1; j < 6; ++j)
                        out[idx++] = Bacc[(k * 6 + i) * NFILT + (m * 6 + j)];
        }
    }
}

extern "C" void kernel_launch(void* const* d_in, const int* in_sizes, int n_in,
                              void* d_out, int out_size, void* d_ws, size_t ws_size,
                              hipStream_t stream) {
    const float* x_hat = (const float*)d_in[0];   // [512,512,2]
    const float* phi_r = (const float*)d_in[1];   // [512,512]
    const float* phi_i = (const float*)d_in[2];   // [512,512]
    const float* psi_r = (const float*)d_in[3];   // [8,6,512,512]
    const float* psi_i = (const float*)d_in[4];   // [8,6,512,512]
    float* out = (float*)d_out;

    // Workspace layout (floats):
    //   [0, 2304)                      Bacc (48x48 accumulator)
    //   [2304, 2352)                   a_ws
    //   [2352, 2353)                   s0_ws
    //   [4096, 4096 + 48*NN*2)         complex intermediate field (100 MB)
    //   [.., .. + 48*NN)               power (50 MB)
    float* ws    = (float*)d_ws;
    float* Bacc  = ws;
    float* a_ws  = ws + 2304;
    float* s0_ws = ws + 2352;
    float2* buf  = (float2*)(ws + 4096);
    float* power = ws + 4096 + (size_t)NFILT * NN * 2;

    k_zero<<<(NFILT * NFILT + 255) / 256, 256, 0, stream>>>(Bacc, NFILT * NFILT);
    k_row_ifft_mul<<<dim3(NSZ, NFILT), 256, 0, stream>>>(x_hat, psi_r, psi_i, buf);
    k_col_ifft_mod_fft<<<dim3(NSZ, NFILT), 256, 0, stream>>>(buf);
    k_row_fft_power<<<dim3(NSZ, NFILT), 256, 0, stream>>>(buf, power);
    k_s0<<<1, 256, 0, stream>>>(x_hat, phi_r, phi_i, s0_ws);
    k_a<<<NFILT, 256, 0, stream>>>(power, phi_r, phi_i, a_ws);
    k_wmma_B<<<dim3(9, 256), 32, 0, stream>>>(power, psi_r, psi_i, Bacc);
    k_assemble<<<1, 1, 0, stream>>>(s0_ws, a_ws, Bacc, out);
}